// GNN_layer_65369402245876
// MI455X (gfx1250) — compile-verified
//
#include <hip/hip_runtime.h>
#include <stdint.h>

typedef float v2f __attribute__((ext_vector_type(2)));
typedef float v8f __attribute__((ext_vector_type(8)));
typedef unsigned int u32x4 __attribute__((ext_vector_type(4)));
typedef int i32x4 __attribute__((ext_vector_type(4)));
typedef int i32x8 __attribute__((ext_vector_type(8)));

#define WMMA4(a, b, c) \
    __builtin_amdgcn_wmma_f32_16x16x4_f32(false, (a), false, (b), (short)0, (c), false, false)

namespace {
enum { kB = 2, kN = 512, kD = 768, kS = 64, kM = 128 };
}

// ---------------------------------------------------------------------------
// Tensor Data Mover: 1D contiguous DMA of `nelem` f32 from global -> LDS.
// D# packed per CDNA5 ISA §8.3/8.4: group0 = {count=1 | lds_addr | global_addr
// | type=2}, group1 = {data_size=4B, tensor_dim0=tile_dim0=nelem,
// tensor_dim1=1, tile_dim1/2=0 (1D), dim0_stride=nelem}. Groups 2/3 zero.
// Descriptor dwords forced into SGPRs via readfirstlane (TDM reads SGPR
// groups). Per-wave issue; completion via s_wait_tensorcnt.
// NOTE: this toolchain exposes the 6-arg builtin (extra int32x8 group before
// the cache-policy immediate) — pass zeros for the unused group.
// ---------------------------------------------------------------------------
__device__ __forceinline__ void tdm_load_1d(uint32_t lds_byte_addr,
                                            const void* gptr, uint32_t nelem) {
  const uint64_t ga = (uint64_t)(uintptr_t)gptr;
  uint32_t g0a = 1u;                                   // count=1, user mode
  uint32_t g0b = lds_byte_addr;                        // lds_addr (bytes)
  uint32_t g0c = (uint32_t)ga;                         // global_addr[31:0]
  uint32_t g0d = (uint32_t)((ga >> 32) & 0x01FFFFFFull) | (2u << 30); // [56:32]|type=2

  u32x4 g0;
  g0.x = (uint32_t)__builtin_amdgcn_readfirstlane((int)g0a);
  g0.y = (uint32_t)__builtin_amdgcn_readfirstlane((int)g0b);
  g0.z = (uint32_t)__builtin_amdgcn_readfirstlane((int)g0c);
  g0.w = (uint32_t)__builtin_amdgcn_readfirstlane((int)g0d);

  const uint32_t n = nelem;
  i32x8 g1;
  g1[0] = (int)(2u << 16);                  // data_size = 2 (4 bytes)
  g1[1] = (int)((n & 0xFFFFu) << 16);       // tensor_dim0[15:0] -> bits[31:16]
  g1[2] = (int)(((n >> 16) & 0xFFFFu)       // tensor_dim0[31:16]
                | (1u << 16));              // tensor_dim1 = 1
  g1[3] = (int)((n & 0xFFFFu) << 16);       // tile_dim0 = n (bits 127:112)
  g1[4] = 0;                                // tile_dim1 = 0, tile_dim2 = 0 (1D)
  g1[5] = (int)n;                           // tensor_dim0_stride[31:0]
  g1[6] = 0;
  g1[7] = 0;
#pragma unroll
  for (int t = 0; t < 8; ++t) g1[t] = __builtin_amdgcn_readfirstlane(g1[t]);

  const i32x4 gz4 = {0, 0, 0, 0};
  const i32x8 gz8 = {0, 0, 0, 0, 0, 0, 0, 0};
  __builtin_amdgcn_tensor_load_to_lds(g0, g1, gz4, gz4, gz8, 0);
}

// ---------------------------------------------------------------------------
// Generic fp32 WMMA GEMM:
//   C[M,N] = concat(A1[M,K1], A2[M,K2]) @ Bm[(K1+K2),N] + bias[N] (+ resid[M,N])
// Block = 256 threads = 8 waves. Wave w computes rows [bx*128 + w*16, +16),
// cols [by*64, +64) as 4 accumulators of v8f via v_wmma_f32_16x16x4_f32.
// B staged in LDS in 64-row K-chunks (shared by all 8 waves).
// Requires: M % 128 == 0, N % 64 == 0, K1 % 4 == 0, K2 % 4 == 0.
// ---------------------------------------------------------------------------
template <bool HAS_A2, bool HAS_BIAS, bool HAS_RESID>
__global__ __launch_bounds__(256) void gemm_wmma_f32(
    const float* __restrict__ A1, int K1,
    const float* __restrict__ A2, int K2,
    const float* __restrict__ Bm,
    const float* __restrict__ bias,
    const float* __restrict__ resid,
    float* __restrict__ C, int M, int N) {
  __shared__ float Bs[64 * 64];  // 16 KB K-chunk x N-tile of B

  const int tid   = threadIdx.x;
  const int wave  = tid >> 5;
  const int lane  = tid & 31;
  const int lhalf = lane >> 4;   // half-wave: selects K pair / +8 output rows
  const int lmod  = lane & 15;
  const int m0    = blockIdx.x * 128 + wave * 16;
  const int n0    = blockIdx.y * 64;
  const int row   = m0 + lmod;   // A-fragment row for this lane
  const int K     = HAS_A2 ? (K1 + K2) : K1;

  v8f acc[4] = {};

  for (int kc0 = 0; kc0 < K; kc0 += 64) {
    const int kchunk = (K - kc0 < 64) ? (K - kc0) : 64;
    __syncthreads();
    // cooperative stage of B[kc0 .. kc0+kchunk) x [n0, n0+64) into LDS (float4)
    for (int idx = tid; idx < kchunk * 16; idx += 256) {
      const int kr = idx >> 4;
      const int c4 = idx & 15;
      ((float4*)(Bs + kr * 64))[c4] =
          *((const float4*)(Bm + (size_t)(kc0 + kr) * N + n0) + c4);
    }
    __syncthreads();

    for (int kk = 0; kk < kchunk; kk += 4) {
      const int kg = kc0 + kk + 2 * lhalf;  // global K of this lane's A pair
      v2f a;
      if (!HAS_A2 || kg < K1) {
        const float* ap = A1 + (size_t)row * K1 + kg;
        a.x = ap[0];
        a.y = ap[1];
      } else {
        const float* ap = A2 + (size_t)row * K2 + (kg - K1);
        a.x = ap[0];
        a.y = ap[1];
      }
      const int kl = kk + 2 * lhalf;
#pragma unroll
      for (int nt = 0; nt < 4; ++nt) {
        v2f bfr;
        bfr.x = Bs[(kl)*64 + nt * 16 + lmod];
        bfr.y = Bs[(kl + 1) * 64 + nt * 16 + lmod];
        acc[nt] = WMMA4(a, bfr, acc[nt]);
      }
    }
  }

  // Epilogue: D layout — lane holds col = n0+nt*16+lmod, VGPR r -> row m0+r+8*lhalf
#pragma unroll
  for (int nt = 0; nt < 4; ++nt) {
    const int ocol = n0 + nt * 16 + lmod;
    const float bv = HAS_BIAS ? bias[ocol] : 0.0f;
#pragma unroll
    for (int r = 0; r < 8; ++r) {
      const int orow = m0 + r + 8 * lhalf;
      float v = acc[nt][r] + bv;
      if (HAS_RESID) v += resid[(size_t)orow * N + ocol];
      C[(size_t)orow * N + ocol] = v;
    }
  }
}

// ---------------------------------------------------------------------------
// message[b,i,c] = (1/N) * sum_j edge[b,i,j,c] * s[b,j,c]
// One 256-thread block per (b,i); thread t: channel c = t&63, j-group t>>6.
// Bound by the 134 MB edge read -> prefetch ahead, 4x unroll.
// ---------------------------------------------------------------------------
__global__ __launch_bounds__(256) void message_reduce(
    const float* __restrict__ edge, const float* __restrict__ s,
    float* __restrict__ msg) {
  __shared__ float red[256];
  const int bi = blockIdx.x;  // b*N + i
  const int b  = bi / kN;
  const float* erow = edge + (size_t)bi * kN * kS;
  const float* sb   = s + (size_t)b * kN * kS;
  const int c  = threadIdx.x & (kS - 1);
  const int jg = threadIdx.x >> 6;
  float acc = 0.0f;
#pragma unroll 4
  for (int j = jg; j < kN; j += 4) {
    __builtin_prefetch(erow + (size_t)(j + 32) * kS + c, 0, 1);
    acc += erow[(size_t)j * kS + c] * sb[(size_t)j * kS + c];
  }
  red[threadIdx.x] = acc;
  __syncthreads();
  if (threadIdx.x < kS) {
    const float v = red[threadIdx.x] + red[threadIdx.x + 64] +
                    red[threadIdx.x + 128] + red[threadIdx.x + 192];
    msg[(size_t)bi * kS + c] = v * (1.0f / kN);
  }
}

// ---------------------------------------------------------------------------
// Fused edge update:
//   out[b,i,j,:] = e + 0.2*(e>0) * (e @ We_edge + Aself[b,i,:] + Bn[b,j,:])
// Block = 256 threads = 8 waves; block covers (b, i, 128 consecutive j).
// Each wave's 16x64 edge sub-tile (contiguous 4 KB) is DMA'd into LDS by the
// Tensor Data Mover (tensor_load_to_lds + s_wait_tensorcnt); We_edge (16 KB)
// staged cooperatively. Each wave runs a 16x64 WMMA tile (64 v_wmma).
// ---------------------------------------------------------------------------
__global__ __launch_bounds__(256) void edge_update_kernel(
    const float* __restrict__ edge,
    const float* __restrict__ We_edge,  // [64,64] row-major (K x N)
    const float* __restrict__ Aself,    // [B,N,64] = s2@We_self + be
    const float* __restrict__ Bn,       // [B,N,64] = s2@We_neigh
    float* __restrict__ out) {
  __shared__ float Wl[64 * 64];   // 16 KB We_edge
  __shared__ float Et[128 * 64];  // 32 KB edge tile (WMMA A + mask/residual)
  __shared__ float As[64];

  const int tid   = threadIdx.x;
  const int wave  = tid >> 5;
  const int lane  = tid & 31;
  const int lhalf = lane >> 4;
  const int lmod  = lane & 15;
  const int b  = blockIdx.z;
  const int i  = blockIdx.y;
  const int j0 = blockIdx.x * 128;

  const size_t row0 = (size_t)(b * kN + i) * kN + j0;  // flat edge-row index
  const float* E = edge + row0 * kS;

  // Per-wave TDM: DMA this wave's 16x64 f32 tile (contiguous 4 KB) into LDS.
  {
    const float* src = E + (size_t)wave * 16 * kS;
    const uint32_t lds_addr = (uint32_t)(uintptr_t)(Et + wave * 16 * kS);
    tdm_load_1d(lds_addr, src, 16 * kS);
  }

  // Cooperative stage of We_edge and Aself while the DMA is in flight.
  for (int idx = tid; idx < 64 * 16; idx += 256)
    ((float4*)Wl)[idx] = ((const float4*)We_edge)[idx];
  if (tid < 64) As[tid] = Aself[(size_t)(b * kN + i) * kS + tid];

  __builtin_amdgcn_s_wait_tensorcnt(0);
  __syncthreads();

  const int r0 = wave * 16;          // this wave's local j-row base
  const float* Ew = Et + r0 * 64;
  v8f acc[4] = {};

#pragma unroll
  for (int kk = 0; kk < 64; kk += 4) {
    const int kl = kk + 2 * lhalf;
    v2f a;
    a.x = Ew[lmod * 64 + kl];        // A frag: lane lmod = row, lhalf = K pair
    a.y = Ew[lmod * 64 + kl + 1];
#pragma unroll
    for (int nt = 0; nt < 4; ++nt) {
      v2f bfr;
      bfr.x = Wl[(kl)*64 + nt * 16 + lmod];
      bfr.y = Wl[(kl + 1) * 64 + nt * 16 + lmod];
      acc[nt] = WMMA4(a, bfr, acc[nt]);
    }
  }

  float* O = out + row0 * kS;
  const float* Bnb = Bn + (size_t)(b * kN + j0) * kS;
#pragma unroll
  for (int nt = 0; nt < 4; ++nt) {
    const int ocol = nt * 16 + lmod;
    const float aself = As[ocol];
#pragma unroll
    for (int r = 0; r < 8; ++r) {
      const int rl = r0 + r + 8 * lhalf;  // local j row
      const float e = Et[rl * 64 + ocol];
      const float v = acc[nt][r] + aself + Bnb[(size_t)rl * kS + ocol];
      const float m = (e > 0.0f) ? 0.2f : 0.0f;
      O[(size_t)rl * kS + ocol] = e + v * m;
    }
  }
}

// ---------------------------------------------------------------------------
extern "C" void kernel_launch(void* const* d_in, const int* in_sizes, int n_in,
                              void* d_out, int out_size, void* d_ws, size_t ws_size,
                              hipStream_t stream) {
  (void)in_sizes; (void)n_in; (void)out_size; (void)ws_size;
  const float* node = (const float*)d_in[0];  // [B,N,D]
  const float* edge = (const float*)d_in[1];  // [B,N,N,S]
  const float* Ws   = (const float*)d_in[2];  // [D,S]
  const float* bs   = (const float*)d_in[3];  // [S]
  const float* Wm   = (const float*)d_in[4];  // [S,M]
  const float* bm   = (const float*)d_in[5];  // [M]
  const float* Wu   = (const float*)d_in[6];  // [D+M,D]
  const float* bu   = (const float*)d_in[7];  // [D]
  const float* We   = (const float*)d_in[8];  // [3S,S]
  const float* be   = (const float*)d_in[9];  // [S]

  float* node_out = (float*)d_out;                         // [B,N,D]
  float* edge_out = (float*)d_out + (size_t)kB * kN * kD;  // [B,N,N,S]

  // workspace layout (floats)
  float* ws    = (float*)d_ws;
  float* s     = ws;                 // [B*N, S]   65536
  float* msg   = s + 65536;          // [B*N, S]   65536
  float* h     = msg + 65536;        // [B*N, M]   131072
  float* s2    = h + 131072;         // [B*N, S]   65536
  float* Aself = s2 + 65536;         // [B*N, S]   65536
  float* Bnbuf = Aself + 65536;      // [B*N, S]   65536

  const int BN = kB * kN;  // 1024 rows for all node-side GEMMs
  const dim3 blk(256);

  // 1) s = node @ Ws + bs                       [1024,768]@[768,64]
  gemm_wmma_f32<false, true, false><<<dim3(BN / 128, kS / 64), blk, 0, stream>>>(
      node, kD, nullptr, 0, Ws, bs, nullptr, s, BN, kS);

  // 2) msg[b,i,:] = mean_j edge[b,i,j,:] * s[b,j,:]
  message_reduce<<<dim3(BN), blk, 0, stream>>>(edge, s, msg);

  // 3) h = msg @ Wm + bm                        [1024,64]@[64,128]
  gemm_wmma_f32<false, true, false><<<dim3(BN / 128, kM / 64), blk, 0, stream>>>(
      msg, kS, nullptr, 0, Wm, bm, nullptr, h, BN, kM);

  // 4) node_out = node + [node | h] @ Wu + bu   [1024,896]@[896,768]
  gemm_wmma_f32<true, true, true><<<dim3(BN / 128, kD / 64), blk, 0, stream>>>(
      node, kD, h, kM, Wu, bu, node, node_out, BN, kD);

  // 5) s2 = node_out @ Ws + bs
  gemm_wmma_f32<false, true, false><<<dim3(BN / 128, kS / 64), blk, 0, stream>>>(
      node_out, kD, nullptr, 0, Ws, bs, nullptr, s2, BN, kS);

  // 6) Aself = s2 @ We_self + be   (We rows [0,64))
  gemm_wmma_f32<false, true, false><<<dim3(BN / 128, kS / 64), blk, 0, stream>>>(
      s2, kS, nullptr, 0, We, be, nullptr, Aself, BN, kS);

  // 7) Bn = s2 @ We_neigh          (We rows [64,128))
  gemm_wmma_f32<false, false, false><<<dim3(BN / 128, kS / 64), blk, 0, stream>>>(
      s2, kS, nullptr, 0, We + 64 * kS, nullptr, nullptr, Bnbuf, BN, kS);

  // 8) edge_out = edge + 0.2*(edge>0)*(edge@We_edge + Aself_i + Bn_j)
  //    We_edge = We rows [128,192)
  edge_update_kernel<<<dim3(kN / 128, kN, kB), blk, 0, stream>>>(
      edge, We + 128 * kS, Aself, Bnbuf, edge_out);
}